// ConsistentHashEmbedder_63788854280735
// MI455X (gfx1250) — compile-verified
//
#include <hip/hip_runtime.h>
#include <hip/hip_bf16.h>
#include <stdint.h>
#include <math.h>

#define NLVL   16
#define LOG2T  19
#define TSZ    (1u << LOG2T)
#define HMASK  (TSZ - 1u)
#define PRIME1 2654435761u
#define NSTAGE 5
#define BLOCK  256

// Staged coarse levels: res = floor(16 * b^l), b = 128^(1/15). These values are
// unambiguous in f32 for l=0..4 and fixed at compile time so LDS can be sized.
#define SRES0 16
#define SRES1 22
#define SRES2 30
#define SRES3 42
#define SRES4 58
#define SENT(r) ((r)*(r) + (r) + 1)            // max dense idx = r*r + r
#define SOFF0 0
#define SOFF1 (SOFF0 + SENT(SRES0))            // 273
#define SOFF2 (SOFF1 + SENT(SRES1))            // 780
#define SOFF3 (SOFF2 + SENT(SRES2))            // 1711
#define SOFF4 (SOFF3 + SENT(SRES3))            // 3518
#define STOTAL (SOFF4 + SENT(SRES4))           // 6941 float2 = 55528 B LDS

struct LevelParam { float inv_gs; uint32_t res; uint32_t hashed; uint32_t pad; };
struct Params { LevelParam lp[NLVL]; };

typedef uint32_t v4u __attribute__((ext_vector_type(4)));
typedef int      v4i __attribute__((ext_vector_type(4)));
typedef int      v8i __attribute__((ext_vector_type(8)));

__device__ __forceinline__ void tdm_load_level(const float* tables, uint32_t lds_byte,
                                               uint32_t level, uint32_t nent) {
#if __has_builtin(__builtin_amdgcn_tensor_load_to_lds)
  uint64_t ga = (uint64_t)(uintptr_t)tables + (uint64_t)level * TSZ * 2u * 4u;
  v4u g0; v8i g1; v4i g2 = {0,0,0,0}; v4i g3 = {0,0,0,0};
  // D# group0: count=1, lds_addr, 57-bit global addr, type=2 ("image")
  g0[0] = 1u;
  g0[1] = lds_byte;
  g0[2] = (uint32_t)ga;
  g0[3] = (uint32_t)((ga >> 32) & 0x01FFFFFFu) | (2u << 30);
  // D# group1: data_size=3 (8B elems), 1-D tile: tensor_dim0 = tile_dim0 = nent
  g1[0] = (int)(3u << 16);                       // data_size=8B; no wg mask/pad/iter
  g1[1] = (int)((nent & 0xFFFFu) << 16);         // tensor_dim0[15:0]
  g1[2] = (int)((nent >> 16) | (1u << 16));      // tensor_dim0[31:16], tensor_dim1=1
  g1[3] = (int)((nent & 0xFFFFu) << 16);         // tile_dim0 = nent (<= 3423)
  g1[4] = 0;                                     // tile_dim1=0 (1-D), tile_dim2=0
  g1[5] = (int)nent;                             // tensor_dim0_stride[31:0]
  g1[6] = 0;
  g1[7] = 0;
#if defined(__clang_major__) && __clang_major__ >= 23
  v8i g4 = {0,0,0,0,0,0,0,0};
  __builtin_amdgcn_tensor_load_to_lds(g0, g1, g2, g3, g4, 0);
#else
  __builtin_amdgcn_tensor_load_to_lds(g0, g1, g2, g3, 0);
#endif
#endif
}

__global__ __launch_bounds__(BLOCK)
void hashgrid_embed_kernel(const float2* __restrict__ x,
                           const float*  __restrict__ tables,
                           float*        __restrict__ out,
                           Params P, int B) {
  __shared__ float2 s_tab[STOTAL];

  // ---- Stage coarse tables (levels 0..4, 54 KB) into LDS ----
#if __has_builtin(__builtin_amdgcn_tensor_load_to_lds)
  if (threadIdx.x < 32) {  // wave 0 only; TDM is a per-wave DMA op
    uint32_t lds0 = (uint32_t)(uintptr_t)&s_tab[0];  // low 32 bits = LDS byte offset
    tdm_load_level(tables, lds0 + SOFF0 * 8u, 0u, SENT(SRES0));
    tdm_load_level(tables, lds0 + SOFF1 * 8u, 1u, SENT(SRES1));
    tdm_load_level(tables, lds0 + SOFF2 * 8u, 2u, SENT(SRES2));
    tdm_load_level(tables, lds0 + SOFF3 * 8u, 3u, SENT(SRES3));
    tdm_load_level(tables, lds0 + SOFF4 * 8u, 4u, SENT(SRES4));
#if __has_builtin(__builtin_amdgcn_s_wait_tensorcnt)
    __builtin_amdgcn_s_wait_tensorcnt(0);
#else
    asm volatile("s_wait_tensorcnt 0" ::: "memory");
#endif
  }
#else
  {
    const int ent[NSTAGE] = {SENT(SRES0), SENT(SRES1), SENT(SRES2), SENT(SRES3), SENT(SRES4)};
    const int off[NSTAGE] = {SOFF0, SOFF1, SOFF2, SOFF3, SOFF4};
    for (int l = 0; l < NSTAGE; ++l) {
      const float2* src = (const float2*)(tables + (size_t)l * TSZ * 2u);
      for (int i = (int)threadIdx.x; i < ent[l]; i += BLOCK)
        s_tab[off[l] + i] = src[i];
    }
  }
#endif
  __syncthreads();

  const int pid = (int)(blockIdx.x * BLOCK + threadIdx.x);
  if (pid >= B) return;

  const float2 p = x[pid];
  float outv[2 * NLVL];

  const uint32_t sres[NSTAGE] = {SRES0, SRES1, SRES2, SRES3, SRES4};
  const uint32_t soff[NSTAGE] = {SOFF0, SOFF1, SOFF2, SOFF3, SOFF4};

#pragma unroll
  for (int l = 0; l < NLVL; ++l) {
    const float    inv = P.lp[l].inv_gs;
    const uint32_t res = P.lp[l].res;

    float t0 = p.x * inv;
    float t1 = p.y * inv;
    float f0 = floorf(t0), f1 = floorf(t1);
    float w0 = t0 - f0,    w1 = t1 - f1;
    uint32_t c0x = (uint32_t)(int)f0, c0y = (uint32_t)(int)f1;
    uint32_t c1x = c0x + 1u,          c1y = c0y + 1u;

    float2 e00, e01, e10, e11;
    if (l < NSTAGE) {
      // LDS-resident dense level: idx = cx*res + cy (matches reference stride bug)
      uint32_t r = sres[l], base = soff[l];
      uint32_t i00 = base + c0x * r + c0y;
      e00 = s_tab[i00];
      e01 = s_tab[i00 + 1u];
      e10 = s_tab[i00 + r];
      e11 = s_tab[i00 + r + 1u];
    } else {
      const float2* tab = (const float2*)tables + (size_t)l * TSZ;
      uint32_t i00, i01, i10, i11;
      if (P.lp[l].hashed) {
        // searchsorted over all vertex hashes is an identity (corner hash always present)
        uint32_t hy0 = c0y * PRIME1, hy1 = c1y * PRIME1;
        i00 = (c0x ^ hy0) & HMASK;
        i01 = (c0x ^ hy1) & HMASK;
        i10 = (c1x ^ hy0) & HMASK;
        i11 = (c1x ^ hy1) & HMASK;
      } else {
        i00 = c0x * res + c0y;
        i01 = i00 + 1u;
        i10 = i00 + res;
        i11 = i10 + 1u;
      }
      e00 = tab[i00]; e01 = tab[i01]; e10 = tab[i10]; e11 = tab[i11];
    }

    float om1 = 1.0f - w1, om0 = 1.0f - w0;
    float cx0 = e00.x * om1 + e01.x * w1;
    float cy0 = e00.y * om1 + e01.y * w1;
    float cx1 = e10.x * om1 + e11.x * w1;
    float cy1 = e10.y * om1 + e11.y * w1;
    outv[2 * l]     = cx0 * om0 + cx1 * w0;
    outv[2 * l + 1] = cy0 * om0 + cy1 * w0;
  }

  // 32 contiguous floats per point -> 8x b128 stores; wave covers contiguous 4 KB
  float4* o4 = (float4*)(out + (size_t)pid * (2 * NLVL));
#pragma unroll
  for (int k = 0; k < (2 * NLVL) / 4; ++k)
    o4[k] = make_float4(outv[4 * k], outv[4 * k + 1], outv[4 * k + 2], outv[4 * k + 3]);
}

extern "C" void kernel_launch(void* const* d_in, const int* in_sizes, int n_in,
                              void* d_out, int out_size, void* d_ws, size_t ws_size,
                              hipStream_t stream) {
  (void)n_in; (void)d_ws; (void)ws_size; (void)out_size;
  const float2* x      = (const float2*)d_in[0];
  const float*  tables = (const float*)d_in[1];
  float*        out    = (float*)d_out;
  const int B = in_sizes[0] / 2;

  // Mirror the reference's float32 resolution computation.
  Params P;
  float b = expf((logf(2048.0f) - logf(16.0f)) / 15.0f);
  for (int l = 0; l < NLVL; ++l) {
    float rf = floorf(16.0f * powf(b, (float)l));
    int res = (int)rf;
    P.lp[l].res    = (uint32_t)res;
    P.lp[l].inv_gs = (float)res / 2048.0f;  // exact binary scale (res * 2^-11)
    long rp1 = (long)res + 1;
    P.lp[l].hashed = (rp1 * rp1 > (long)TSZ) ? 1u : 0u;
    P.lp[l].pad    = 0u;
  }

  dim3 grid((unsigned)((B + BLOCK - 1) / BLOCK));
  hashgrid_embed_kernel<<<grid, dim3(BLOCK), 0, stream>>>(x, tables, out, P, B);
}